// CTDGConv_21492016349930
// MI455X (gfx1250) — compile-verified
//
#include <hip/hip_runtime.h>
#include <math.h>

#define NN 50000
#define NE 800000
#define NODE_DIM 128
#define EDGE_DIM 64
#define TIME_DIM 16
#define OUT_DIM 128
#define IN_DIM 208       // 128 + 64 + 16
#define N_HEADS 4
#define RED_BLOCKS 256

typedef __attribute__((ext_vector_type(2))) float v2f;
typedef __attribute__((ext_vector_type(4))) float v4f;
typedef __attribute__((ext_vector_type(8))) float v8f;

// ---------------------------------------------------------------------------
// Kernel 1: msgs = [x[src], edge_feat, time_enc] @ Wm + bm   (fp32 WMMA)
//           logits = [msgs, x[dst]] @ Wa + ba                (in-block)
// 32 edges per block, 8 waves x 16-column strips, K-step 4.
// ---------------------------------------------------------------------------
#define K1_AS 212            // LDS row stride: 16B-aligned, conflict-free (212*r mod 64 injective)
#define MSG_S 132            // msgs-in-LDS row stride

__global__ __launch_bounds__(256) void k_msg_gemm(
    const float* __restrict__ x, const int* __restrict__ eidx,
    const float* __restrict__ efeat, const float* __restrict__ dt,
    const float* __restrict__ gammas, const float* __restrict__ Wm,
    const float* __restrict__ bm, const float* __restrict__ Wa,
    const float* __restrict__ ba,
    float* __restrict__ msgs, float* __restrict__ logits)
{
  __shared__ float sA[32 * K1_AS];
  __shared__ float sG[TIME_DIM];
  __shared__ int   sSrc[32], sDst[32];

  const int tid = threadIdx.x;
  const int e0  = blockIdx.x * 32;

  if (tid < TIME_DIM) {
    float g = gammas[tid];
    sG[tid] = (g > 20.f) ? g : log1pf(expf(g));   // softplus
  }
  if (tid < 32) { sSrc[tid] = eidx[e0 + tid]; sDst[tid] = eidx[NE + e0 + tid]; }
  __syncthreads();

  // Stage A tile (32 edges x 208 feats) as float4: 32*52 vectors
  for (int idx = tid; idx < 32 * (IN_DIM / 4); idx += 256) {
    const int row = idx / (IN_DIM / 4);
    const int c4  = idx - row * (IN_DIM / 4);
    v4f v;
    if (c4 < NODE_DIM / 4) {                       // x[src] : L2-resident gather
      v = *(const v4f*)(x + (size_t)sSrc[row] * NODE_DIM + c4 * 4);
    } else if (c4 < (NODE_DIM + EDGE_DIM) / 4) {   // edge_feat : one-shot stream -> NT
      v = __builtin_nontemporal_load(
            (const v4f*)(efeat + (size_t)(e0 + row) * EDGE_DIM + (c4 * 4 - NODE_DIM)));
    } else {                                       // time encoding
      float d = dt[e0 + row]; d = d > 0.f ? d : 0.f;
      const int kb = c4 * 4 - NODE_DIM - EDGE_DIM;
      v.x = expf(-sG[kb + 0] * d);
      v.y = expf(-sG[kb + 1] * d);
      v.z = expf(-sG[kb + 2] * d);
      v.w = expf(-sG[kb + 3] * d);
    }
    *(v4f*)(&sA[row * K1_AS + c4 * 4]) = v;
  }
  __syncthreads();

  const int lane   = tid & 31;
  const int wave   = tid >> 5;
  const int nloc   = lane & 15;
  const int n      = wave * 16 + nloc;        // output column
  const int khalf  = (lane >> 4) << 1;        // 0 or 2
  const int rowoff = (lane >> 4) << 3;        // 0 or 8 (C-matrix row offset)

  v8f acc0 = {}, acc1 = {};
  for (int k0 = 0; k0 < IN_DIM; k0 += 4) {
    const int kk = k0 + khalf;
    v2f b;  b.x  = Wm[kk * OUT_DIM + n];            b.y  = Wm[(kk + 1) * OUT_DIM + n];
    v2f a0; a0.x = sA[nloc * K1_AS + kk];           a0.y = sA[nloc * K1_AS + kk + 1];
    v2f a1; a1.x = sA[(16 + nloc) * K1_AS + kk];    a1.y = sA[(16 + nloc) * K1_AS + kk + 1];
    acc0 = __builtin_amdgcn_wmma_f32_16x16x4_f32(false, a0, false, b, (short)0, acc0, false, false);
    acc1 = __builtin_amdgcn_wmma_f32_16x16x4_f32(false, a1, false, b, (short)0, acc1, false, false);
  }

  const float bias = bm[n];
  __syncthreads();   // all waves done reading sA; reuse it for msgs staging
  #pragma unroll
  for (int r = 0; r < 8; ++r) {
    const float v0 = acc0[r] + bias;
    const float v1 = acc1[r] + bias;
    const int row0 = r + rowoff;
    const int row1 = row0 + 16;
    // msgs stream (410 MB, > L2): write non-temporal so x stays L2-resident
    __builtin_nontemporal_store(v0, &msgs[(size_t)(e0 + row0) * OUT_DIM + n]);
    __builtin_nontemporal_store(v1, &msgs[(size_t)(e0 + row1) * OUT_DIM + n]);
    sA[row0 * MSG_S + n] = v0;
    sA[row1 * MSG_S + n] = v1;
  }
  __syncthreads();

  // Attention logits: one (edge, head) per thread (128 active)
  if (tid < 32 * N_HEADS) {
    const int e = tid >> 2, h = tid & 3;
    float acc = ba[h];
    const float* xd = x + (size_t)sDst[e] * NODE_DIM;
    for (int c = 0; c < OUT_DIM; ++c)  acc += sA[e * MSG_S + c] * Wa[c * N_HEADS + h];
    for (int c = 0; c < NODE_DIM; ++c) acc += xd[c] * Wa[(OUT_DIM + c) * N_HEADS + h];
    logits[(size_t)(e0 + e) * N_HEADS + h] = acc;
  }
}

// ---------------------------------------------------------------------------
// Global softmax over dim=0 (all edges), per head: max then sum-exp.
// ---------------------------------------------------------------------------
__global__ __launch_bounds__(256) void k_max_partial(const float* __restrict__ logits,
                                                     float* __restrict__ pmax)
{
  __shared__ float red[256 * N_HEADS];
  const int tid = threadIdx.x;
  float lm[N_HEADS] = {-3.0e38f, -3.0e38f, -3.0e38f, -3.0e38f};
  for (int e = blockIdx.x * blockDim.x + tid; e < NE; e += gridDim.x * blockDim.x) {
    const v4f l = *(const v4f*)(logits + (size_t)e * N_HEADS);
    lm[0] = fmaxf(lm[0], l.x); lm[1] = fmaxf(lm[1], l.y);
    lm[2] = fmaxf(lm[2], l.z); lm[3] = fmaxf(lm[3], l.w);
  }
  #pragma unroll
  for (int h = 0; h < N_HEADS; ++h) red[tid * N_HEADS + h] = lm[h];
  __syncthreads();
  for (int s = 128; s > 0; s >>= 1) {
    if (tid < s)
      for (int h = 0; h < N_HEADS; ++h)
        red[tid * N_HEADS + h] = fmaxf(red[tid * N_HEADS + h], red[(tid + s) * N_HEADS + h]);
    __syncthreads();
  }
  if (tid < N_HEADS) pmax[blockIdx.x * N_HEADS + tid] = red[tid];
}

__global__ void k_max_final(const float* __restrict__ pmax, float* __restrict__ gmax)
{
  if (threadIdx.x < N_HEADS) {
    float m = -3.0e38f;
    for (int b = 0; b < RED_BLOCKS; ++b) m = fmaxf(m, pmax[b * N_HEADS + threadIdx.x]);
    gmax[threadIdx.x] = m;
  }
}

__global__ __launch_bounds__(256) void k_sum_partial(const float* __restrict__ logits,
                                                     const float* __restrict__ gmax,
                                                     float* __restrict__ psum)
{
  __shared__ float red[256 * N_HEADS];
  const int tid = threadIdx.x;
  float m[N_HEADS];
  #pragma unroll
  for (int h = 0; h < N_HEADS; ++h) m[h] = gmax[h];
  float ls[N_HEADS] = {0.f, 0.f, 0.f, 0.f};
  for (int e = blockIdx.x * blockDim.x + tid; e < NE; e += gridDim.x * blockDim.x) {
    const v4f l = *(const v4f*)(logits + (size_t)e * N_HEADS);
    ls[0] += expf(l.x - m[0]); ls[1] += expf(l.y - m[1]);
    ls[2] += expf(l.z - m[2]); ls[3] += expf(l.w - m[3]);
  }
  #pragma unroll
  for (int h = 0; h < N_HEADS; ++h) red[tid * N_HEADS + h] = ls[h];
  __syncthreads();
  for (int s = 128; s > 0; s >>= 1) {
    if (tid < s)
      for (int h = 0; h < N_HEADS; ++h)
        red[tid * N_HEADS + h] += red[(tid + s) * N_HEADS + h];
    __syncthreads();
  }
  if (tid < N_HEADS) psum[blockIdx.x * N_HEADS + tid] = red[tid];
}

__global__ void k_sum_final(const float* __restrict__ psum, float* __restrict__ gsum)
{
  if (threadIdx.x < N_HEADS) {
    float s = 0.f;
    for (int b = 0; b < RED_BLOCKS; ++b) s += psum[b * N_HEADS + threadIdx.x];
    gsum[threadIdx.x] = s;
  }
}

// ---------------------------------------------------------------------------
// Zero-init agg; scatter weighted messages with f32 atomics (one warp/edge).
// ---------------------------------------------------------------------------
__global__ void k_zero(float* __restrict__ p, int n)
{
  int i = blockIdx.x * blockDim.x + threadIdx.x;
  if (i < n) p[i] = 0.f;
}

__global__ __launch_bounds__(256) void k_scatter(
    const float* __restrict__ msgs, const float* __restrict__ logits,
    const int* __restrict__ eidx, const float* __restrict__ gmax,
    const float* __restrict__ gsum, float* __restrict__ agg)
{
  const int lane = threadIdx.x & 31;
  const int e    = blockIdx.x * 8 + (threadIdx.x >> 5);
  if (e >= NE) return;
  float w = 0.f;
  #pragma unroll
  for (int h = 0; h < N_HEADS; ++h)
    w += expf(logits[(size_t)e * N_HEADS + h] - gmax[h]) / gsum[h];
  w *= 0.25f;  // mean over heads
  const int d = eidx[NE + e];
  // msgs read-once stream: non-temporal b128 per lane (warp covers the 512B row)
  const v4f mv = __builtin_nontemporal_load(
      (const v4f*)(msgs + (size_t)e * OUT_DIM + lane * 4));
  float* ag = agg + (size_t)d * OUT_DIM + lane * 4;
  unsafeAtomicAdd(ag + 0, mv.x * w);
  unsafeAtomicAdd(ag + 1, mv.y * w);
  unsafeAtomicAdd(ag + 2, mv.z * w);
  unsafeAtomicAdd(ag + 3, mv.w * w);
}

// ---------------------------------------------------------------------------
// Kernel 5: out = LN(gelu([agg, x] @ Wo + bo)) * g + b   (fp32 WMMA)
// ---------------------------------------------------------------------------
#define K5_AS 260            // 16B-aligned, conflict-free (260*r mod 64 injective)
#define OUT_S 132

__global__ __launch_bounds__(256) void k_out_gemm(
    const float* __restrict__ agg, const float* __restrict__ x,
    const float* __restrict__ Wo, const float* __restrict__ bo,
    const float* __restrict__ ln_g, const float* __restrict__ ln_b,
    float* __restrict__ out)
{
  __shared__ float sA[32 * K5_AS];
  const int tid = threadIdx.x;
  const int r0  = blockIdx.x * 32;

  // Stage [agg | x] tile as float4: 32 rows x 64 vectors
  for (int idx = tid; idx < 32 * 64; idx += 256) {
    const int row = idx >> 6, c4 = idx & 63;
    const int gr = r0 + row;
    v4f v = {0.f, 0.f, 0.f, 0.f};
    if (gr < NN)
      v = (c4 < 32) ? *(const v4f*)(agg + (size_t)gr * OUT_DIM + c4 * 4)
                    : *(const v4f*)(x + (size_t)gr * NODE_DIM + (c4 - 32) * 4);
    *(v4f*)(&sA[row * K5_AS + c4 * 4]) = v;
  }
  __syncthreads();

  const int lane   = tid & 31;
  const int wave   = tid >> 5;
  const int nloc   = lane & 15;
  const int n      = wave * 16 + nloc;
  const int khalf  = (lane >> 4) << 1;
  const int rowoff = (lane >> 4) << 3;

  v8f acc0 = {}, acc1 = {};
  for (int k0 = 0; k0 < 256; k0 += 4) {
    const int kk = k0 + khalf;
    v2f b;  b.x  = Wo[kk * OUT_DIM + n];          b.y  = Wo[(kk + 1) * OUT_DIM + n];
    v2f a0; a0.x = sA[nloc * K5_AS + kk];         a0.y = sA[nloc * K5_AS + kk + 1];
    v2f a1; a1.x = sA[(16 + nloc) * K5_AS + kk];  a1.y = sA[(16 + nloc) * K5_AS + kk + 1];
    acc0 = __builtin_amdgcn_wmma_f32_16x16x4_f32(false, a0, false, b, (short)0, acc0, false, false);
    acc1 = __builtin_amdgcn_wmma_f32_16x16x4_f32(false, a1, false, b, (short)0, acc1, false, false);
  }

  const float bias = bo[n];
  __syncthreads();
  #pragma unroll
  for (int r = 0; r < 8; ++r) {
    float v0 = acc0[r] + bias;
    float v1 = acc1[r] + bias;
    v0 = 0.5f * v0 * (1.f + erff(v0 * 0.70710678118654752f));   // exact GELU
    v1 = 0.5f * v1 * (1.f + erff(v1 * 0.70710678118654752f));
    sA[(r + rowoff) * OUT_S + n]      = v0;
    sA[(r + rowoff + 16) * OUT_S + n] = v1;
  }
  __syncthreads();

  if (tid < 32) {
    const int gr = r0 + tid;
    if (gr < NN) {
      float mu = 0.f;
      for (int c = 0; c < OUT_DIM; ++c) mu += sA[tid * OUT_S + c];
      mu *= (1.f / OUT_DIM);
      float var = 0.f;
      for (int c = 0; c < OUT_DIM; ++c) {
        const float d = sA[tid * OUT_S + c] - mu;
        var += d * d;
      }
      var *= (1.f / OUT_DIM);
      const float inv = 1.f / sqrtf(var + 1e-5f);
      for (int c = 0; c < OUT_DIM; ++c)
        out[(size_t)gr * OUT_DIM + c] =
            (sA[tid * OUT_S + c] - mu) * inv * ln_g[c] + ln_b[c];
    }
  }
}

// ---------------------------------------------------------------------------
extern "C" void kernel_launch(void* const* d_in, const int* in_sizes, int n_in,
                              void* d_out, int out_size, void* d_ws, size_t ws_size,
                              hipStream_t stream)
{
  (void)in_sizes; (void)n_in; (void)out_size; (void)ws_size;
  const float* x      = (const float*)d_in[0];
  const int*   eidx   = (const int*)  d_in[1];
  const float* efeat  = (const float*)d_in[2];
  const float* dt     = (const float*)d_in[3];
  const float* gammas = (const float*)d_in[4];
  const float* Wm     = (const float*)d_in[5];
  const float* bm     = (const float*)d_in[6];
  const float* Wa     = (const float*)d_in[7];
  const float* ba     = (const float*)d_in[8];
  const float* Wo     = (const float*)d_in[9];
  const float* bo     = (const float*)d_in[10];
  const float* ln_g   = (const float*)d_in[11];
  const float* ln_b   = (const float*)d_in[12];
  float* out = (float*)d_out;

  float* ws     = (float*)d_ws;
  float* msgs   = ws;                                  // E * 128
  float* logits = msgs   + (size_t)NE * OUT_DIM;       // E * 4
  float* agg    = logits + (size_t)NE * N_HEADS;       // N * 128
  float* pmax   = agg    + (size_t)NN * OUT_DIM;       // 256 * 4
  float* psum   = pmax   + RED_BLOCKS * N_HEADS;       // 256 * 4
  float* gmax   = psum   + RED_BLOCKS * N_HEADS;       // 4
  float* gsum   = gmax   + N_HEADS;                    // 4

  k_zero<<<(NN * OUT_DIM + 255) / 256, 256, 0, stream>>>(agg, NN * OUT_DIM);
  k_msg_gemm<<<NE / 32, 256, 0, stream>>>(x, eidx, efeat, dt, gammas, Wm, bm, Wa, ba,
                                          msgs, logits);
  k_max_partial<<<RED_BLOCKS, 256, 0, stream>>>(logits, pmax);
  k_max_final<<<1, 32, 0, stream>>>(pmax, gmax);
  k_sum_partial<<<RED_BLOCKS, 256, 0, stream>>>(logits, gmax, psum);
  k_sum_final<<<1, 32, 0, stream>>>(psum, gsum);
  k_scatter<<<NE / 8, 256, 0, stream>>>(msgs, logits, eidx, gmax, gsum, agg);
  k_out_gemm<<<(NN + 31) / 32, 256, 0, stream>>>(agg, x, Wo, bo, ln_g, ln_b, out);
}